// L2MultiheadAttention_12412455486068
// MI455X (gfx1250) — compile-verified
//
#include <hip/hip_runtime.h>
#include <hip/hip_bf16.h>
#include <math.h>

// L2 multihead attention for MI455X (gfx1250), fp32 WMMA path.
// T=1024, N=8, H=8, hd=64, E=512.
// Pipeline (after algebraic reduction  attn@(x@A)@V == attn@(x@(A@V)) ):
//   1) q      = x @ Wq                      (proj_gemm)      [bh][t][d]
//   2) sq     = rowsum(q*q)                 (sq_kernel)
//   3) logits = (2*q qT - sq_t - sq_s)/8    (logits_kernel)  -> attn region of d_out
//   4) softmax over s, in place             (softmax_kernel)
//   5) CH_h   = Wq_h Wq_h^T / 8             (gram_kernel)
//   6) Bcomb  = CH_h @ Wv_h  (laid [m][h*64+d])  (bcomb_kernel)
//   7) XAV    = x @ Bcomb                   (proj_gemm again) [bh][s][d]
//   8) PXAV   = attn @ XAV  per (b,h)       (pxav_kernel)     [t*8+b][h*64+d]
//   9) out    = PXAV @ Wo^T + b             (out_kernel)      -> out region of d_out

typedef __attribute__((ext_vector_type(2))) float v2f;
typedef __attribute__((ext_vector_type(8))) float v8f;

#define Tn   1024
#define Nb   8
#define Hh   8
#define HD   64
#define Ee   512
#define BH   (Nb*Hh)      // 64

__device__ __forceinline__ v8f wmma4(v2f a, v2f b, v8f c) {
  // D = A(16x4,f32) * B(4x16,f32) + C(16x16,f32)
  return __builtin_amdgcn_wmma_f32_16x16x4_f32(
      /*neg_a=*/false, a, /*neg_b=*/false, b,
      /*c_mod=*/(short)0, c, /*reuse_a=*/false, /*reuse_b=*/false);
}

// ---------------------------------------------------------------------------
// GEMM: rows M=8192 over (t*8+b), A = X [8192,512] row-major,
//       B = W [512,512] row-major (k,n with n = h*64+d),
//       D written to [bh][t][d] layout (bh = b*8+h).
// grid: 16384 tiles / 8 waves = 2048 blocks of 256.
__global__ void proj_gemm_kernel(const float* __restrict__ X,
                                 const float* __restrict__ W,
                                 float* __restrict__ Q) {
  const int wave = threadIdx.x >> 5;
  const int lane = threadIdx.x & 31;
  const int half = lane >> 4;
  const int lrow = lane & 15;
  const int tile = blockIdx.x * 8 + wave;
  const int tm = tile >> 5;             // 512 M-tiles
  const int tn = tile & 31;             // 32  N-tiles
  const int m0 = tm * 16, n0 = tn * 16;

  const float* __restrict__ arow = X + (m0 + lrow) * Ee;
  v8f c = {};
  for (int k = 0; k < Ee; k += 4) {
    __builtin_prefetch(arow + k + 64, 0, 1);
    v2f a, b;
    a.x = arow[k + 2 * half];
    a.y = arow[k + 2 * half + 1];
    b.x = W[(k + 2 * half) * Ee + n0 + lrow];
    b.y = W[(k + 2 * half + 1) * Ee + n0 + lrow];
    c = wmma4(a, b, c);
  }
  const int gn = n0 + lrow;
  const int h = gn >> 6, d = gn & 63;
#pragma unroll
  for (int r = 0; r < 8; ++r) {
    const int gm = m0 + r + 8 * half;        // = t*8 + b
    const int t = gm >> 3, bb = gm & 7;
    Q[(((bb * Hh + h) * Tn) + t) * HD + d] = c[r];
  }
}

// ---------------------------------------------------------------------------
// sq[bh*1024+t] = sum_d q^2 ; 65536 threads.
__global__ void sq_kernel(const float* __restrict__ Q, float* __restrict__ SQ) {
  const int idx = blockIdx.x * blockDim.x + threadIdx.x;   // bh*1024 + t
  const float* __restrict__ row = Q + idx * HD;
  float s = 0.f;
#pragma unroll 8
  for (int d = 0; d < HD; ++d) s += row[d] * row[d];
  SQ[idx] = s;
}

// ---------------------------------------------------------------------------
// logits[t,s,b,h] = (2*dot - sq_t - sq_s) * 0.125, per (b,h) GEMM 1024x1024x64.
// tiles/bh = 64*64 = 4096; total 262144 waves -> 32768 blocks.
__global__ void logits_kernel(const float* __restrict__ Q,
                              const float* __restrict__ SQ,
                              float* __restrict__ ATT) {
  const int wave = threadIdx.x >> 5;
  const int lane = threadIdx.x & 31;
  const int half = lane >> 4;
  const int lrow = lane & 15;
  const int gtile = blockIdx.x * 8 + wave;
  const int bh = gtile >> 12;
  const int tl = gtile & 4095;
  const int tm = tl >> 6, tn = tl & 63;
  const int m0 = tm * 16, n0 = tn * 16;

  const float* __restrict__ Qbh = Q + bh * Tn * HD;
  const float* __restrict__ ar = Qbh + (m0 + lrow) * HD;
  const float* __restrict__ br = Qbh + (n0 + lrow) * HD;
  v8f c = {};
  for (int k = 0; k < HD; k += 4) {
    v2f a, b;
    a.x = ar[k + 2 * half];
    a.y = ar[k + 2 * half + 1];
    b.x = br[k + 2 * half];
    b.y = br[k + 2 * half + 1];
    c = wmma4(a, b, c);
  }
  const int s = n0 + lrow;
  const float sqs = SQ[bh * Tn + s];
#pragma unroll
  for (int r = 0; r < 8; ++r) {
    const int t = m0 + r + 8 * half;
    const float logit = (2.f * c[r] - SQ[bh * Tn + t] - sqs) * 0.125f;
    ATT[t * (Tn * BH) + s * BH + bh] = logit;   // ((t*S+s)*N+b)*H+h
  }
}

// ---------------------------------------------------------------------------
// Softmax over s (1024) per row (t,bh), in place. One wave per row,
// 32 elements per lane. 65536 rows -> 8192 blocks of 8 waves.
__global__ void softmax_kernel(float* __restrict__ ATT) {
  const int wave = threadIdx.x >> 5;
  const int lane = threadIdx.x & 31;
  const int row = blockIdx.x * 8 + wave;       // t*64 + bh
  const int t = row >> 6, bh = row & 63;
  float* __restrict__ base = ATT + t * (Tn * BH) + bh;

  float v[32];
  float m = -INFINITY;
#pragma unroll
  for (int i = 0; i < 32; ++i) {
    v[i] = base[(lane + 32 * i) * BH];
    m = fmaxf(m, v[i]);
  }
#pragma unroll
  for (int off = 16; off; off >>= 1) m = fmaxf(m, __shfl_xor(m, off, 32));
  float sum = 0.f;
#pragma unroll
  for (int i = 0; i < 32; ++i) {
    v[i] = __expf(v[i] - m);
    sum += v[i];
  }
#pragma unroll
  for (int off = 16; off; off >>= 1) sum += __shfl_xor(sum, off, 32);
  const float inv = 1.0f / sum;
#pragma unroll
  for (int i = 0; i < 32; ++i) base[(lane + 32 * i) * BH] = v[i] * inv;
}

// ---------------------------------------------------------------------------
// CH[h][m][n] = sum_d Wq[m,h,d]*Wq[n,h,d] * 0.125 ; per h: 512x512x64.
// tiles/h = 32*32 = 1024; total 8192 waves -> 1024 blocks.
__global__ void gram_kernel(const float* __restrict__ WQ, float* __restrict__ CH) {
  const int wave = threadIdx.x >> 5;
  const int lane = threadIdx.x & 31;
  const int half = lane >> 4;
  const int lrow = lane & 15;
  const int gtile = blockIdx.x * 8 + wave;
  const int h = gtile >> 10;
  const int tl = gtile & 1023;
  const int tm = tl >> 5, tn = tl & 31;
  const int m0 = tm * 16, n0 = tn * 16;

  const float* __restrict__ ar = WQ + (m0 + lrow) * Ee + h * HD;
  const float* __restrict__ br = WQ + (n0 + lrow) * Ee + h * HD;
  v8f c = {};
  for (int k = 0; k < HD; k += 4) {
    v2f a, b;
    a.x = ar[k + 2 * half];
    a.y = ar[k + 2 * half + 1];
    b.x = br[k + 2 * half];
    b.y = br[k + 2 * half + 1];
    c = wmma4(a, b, c);
  }
#pragma unroll
  for (int r = 0; r < 8; ++r)
    CH[h * Ee * Ee + (m0 + r + 8 * half) * Ee + n0 + lrow] = c[r] * 0.125f;
}

// ---------------------------------------------------------------------------
// Bcomb[m][h*64+d] = sum_n CH[h][m][n] * Wv[n,h,d] ; per h: 512x64x512.
// tiles/h = 32*4 = 128; total 1024 waves -> 128 blocks.
__global__ void bcomb_kernel(const float* __restrict__ CH,
                             const float* __restrict__ WV,
                             float* __restrict__ BC) {
  const int wave = threadIdx.x >> 5;
  const int lane = threadIdx.x & 31;
  const int half = lane >> 4;
  const int lrow = lane & 15;
  const int gtile = blockIdx.x * 8 + wave;
  const int h = gtile >> 7;
  const int tl = gtile & 127;
  const int tm = tl >> 2, tn = tl & 3;
  const int m0 = tm * 16, n0 = tn * 16;

  const float* __restrict__ ar = CH + h * Ee * Ee + (m0 + lrow) * Ee;
  v8f c = {};
  for (int k = 0; k < Ee; k += 4) {
    v2f a, b;
    a.x = ar[k + 2 * half];
    a.y = ar[k + 2 * half + 1];
    b.x = WV[(k + 2 * half) * Ee + h * HD + n0 + lrow];
    b.y = WV[(k + 2 * half + 1) * Ee + h * HD + n0 + lrow];
    c = wmma4(a, b, c);
  }
#pragma unroll
  for (int r = 0; r < 8; ++r)
    BC[(m0 + r + 8 * half) * Ee + h * HD + n0 + lrow] = c[r];
}

// ---------------------------------------------------------------------------
// PXAV: per (b,h): [1024 x 64] = attn[t,s] (stride-64 rows) @ XAV[bh][s][d].
// Output layout [(t*8+b)*512 + h*64 + d]. tiles/bh = 64*4 = 256; 2048 blocks.
__global__ void pxav_kernel(const float* __restrict__ ATT,
                            const float* __restrict__ XAV,
                            float* __restrict__ PX) {
  const int wave = threadIdx.x >> 5;
  const int lane = threadIdx.x & 31;
  const int half = lane >> 4;
  const int lrow = lane & 15;
  const int gtile = blockIdx.x * 8 + wave;
  const int bh = gtile >> 8;
  const int tl = gtile & 255;
  const int tm = tl >> 2, tn = tl & 3;
  const int m0 = tm * 16, n0 = tn * 16;

  const float* __restrict__ arow = ATT + (m0 + lrow) * (Tn * BH) + bh;
  const float* __restrict__ Xbh = XAV + bh * Tn * HD;
  v8f c = {};
  for (int k = 0; k < Tn; k += 4) {
    v2f a, b;
    a.x = arow[(k + 2 * half) * BH];
    a.y = arow[(k + 2 * half + 1) * BH];
    b.x = Xbh[(k + 2 * half) * HD + n0 + lrow];
    b.y = Xbh[(k + 2 * half + 1) * HD + n0 + lrow];
    c = wmma4(a, b, c);
  }
  const int bb = bh >> 3, h = bh & 7;
  const int d = n0 + lrow;
#pragma unroll
  for (int r = 0; r < 8; ++r) {
    const int t = m0 + r + 8 * half;
    PX[(t * Nb + bb) * Ee + h * HD + d] = c[r];
  }
}

// ---------------------------------------------------------------------------
// out = PX [8192,512] @ out_w^T + out_b. B[k][n] = OW[n*512+k]. 2048 blocks.
__global__ void out_kernel(const float* __restrict__ PX,
                           const float* __restrict__ OW,
                           const float* __restrict__ OB,
                           float* __restrict__ OUT) {
  const int wave = threadIdx.x >> 5;
  const int lane = threadIdx.x & 31;
  const int half = lane >> 4;
  const int lrow = lane & 15;
  const int tile = blockIdx.x * 8 + wave;
  const int tm = tile >> 5, tn = tile & 31;
  const int m0 = tm * 16, n0 = tn * 16;

  const float* __restrict__ arow = PX + (m0 + lrow) * Ee;
  const float* __restrict__ brow = OW + (n0 + lrow) * Ee;
  v8f c = {};
  for (int k = 0; k < Ee; k += 4) {
    __builtin_prefetch(arow + k + 64, 0, 1);
    v2f a, b;
    a.x = arow[k + 2 * half];
    a.y = arow[k + 2 * half + 1];
    b.x = brow[k + 2 * half];
    b.y = brow[k + 2 * half + 1];
    c = wmma4(a, b, c);
  }
  const float bias = OB[n0 + lrow];
#pragma unroll
  for (int r = 0; r < 8; ++r)
    OUT[(m0 + r + 8 * half) * Ee + n0 + lrow] = c[r] + bias;
}

// ---------------------------------------------------------------------------
extern "C" void kernel_launch(void* const* d_in, const int* in_sizes, int n_in,
                              void* d_out, int out_size, void* d_ws, size_t ws_size,
                              hipStream_t stream) {
  const float* x    = (const float*)d_in[0];   // [T,N,E]
  const float* wq   = (const float*)d_in[1];   // [E,H,hd] == [E,E] row-major
  const float* wv   = (const float*)d_in[2];   // [E,H,hd]
  const float* ow   = (const float*)d_in[3];   // [E,E]
  const float* ob   = (const float*)d_in[4];   // [E]

  float* out_p  = (float*)d_out;                       // [T,N,E] = 4,194,304 floats
  float* attn_p = (float*)d_out + Tn * Nb * Ee;        // [T,S,N,H] = 67,108,864 floats

  // Workspace layout (floats). q region is reused for PXAV (q dead after logits).
  float* ws = (float*)d_ws;
  float* q_ws  = ws;                         // 4,194,304  ([bh][t][d]) ; later PXAV
  float* sq_ws = q_ws + 4194304;             //    65,536
  float* ch_ws = sq_ws + 65536;              // 2,097,152  ([H][E][E])
  float* bc_ws = ch_ws + 2097152;            //   262,144  ([E][E])
  float* xav_ws = bc_ws + 262144;            // 4,194,304  ([bh][s][d])
  // total: 10,813,440 floats = 43.25 MB <= ws_size (assumed)

  dim3 blk(256);

  // 1) q = x @ Wq
  proj_gemm_kernel<<<2048, blk, 0, stream>>>(x, wq, q_ws);
  // 2) sq
  sq_kernel<<<256, blk, 0, stream>>>(q_ws, sq_ws);
  // 3) logits -> attn region
  logits_kernel<<<32768, blk, 0, stream>>>(q_ws, sq_ws, attn_p);
  // 4) softmax in place
  softmax_kernel<<<8192, blk, 0, stream>>>(attn_p);
  // 5) per-head Gram of Wq (scaled)
  gram_kernel<<<1024, blk, 0, stream>>>(wq, ch_ws);
  // 6) Bcomb = CH_h @ Wv_h
  bcomb_kernel<<<128, blk, 0, stream>>>(ch_ws, wv, bc_ws);
  // 7) XAV = x @ Bcomb
  proj_gemm_kernel<<<2048, blk, 0, stream>>>(x, bc_ws, xav_ws);
  // 8) PXAV = attn @ XAV  (writes into q region)
  pxav_kernel<<<2048, blk, 0, stream>>>(attn_p, xav_ws, q_ws);
  // 9) out = PXAV @ Wo^T + b
  out_kernel<<<2048, blk, 0, stream>>>(q_ws, ow, ob, out_p);
}